// EquivariantProductBasisBlock_19653770347273
// MI455X (gfx1250) — compile-verified
//
#include <hip/hip_runtime.h>

typedef float v2f __attribute__((ext_vector_type(2)));
typedef float v8f __attribute__((ext_vector_type(8)));

#define NNODES 2048
#define C 128
#define DIM 9
#define NP2 45
#define NP3 165
#define K3P 168   // 165 padded to mult of 4
#define K2P 48    // 45 -> 48
#define K1P 12    // 9 -> 12
#define U3_ROWS 64   // 63 real rows (9 lm * 7 k)
#define U2_ROWS 32   // 27 real rows (9 lm * 3 k)
#define U1_ROWS 16   // 9 real rows
#define U2_OFF (U3_ROWS * K3P)            // 10752
#define U1_OFF (U2_OFF + U2_ROWS * K2P)   // 12288
#define U_TOTAL (U1_OFF + U1_ROWS * K1P)  // 12480 floats

// LDS layout (float offsets), padded row stride to dodge bank conflicts
#define CSTR 132
#define MROWS 228           // 168 (nu3) + 48 (nu2) + 12 (nu1)
#define YROWS 112           // 64 + 32 + 16
#define LDS_XT 0                          // [9][CSTR]
#define LDS_M  (LDS_XT + DIM * CSTR)      // [228][CSTR]
#define LDS_Y  (LDS_M + MROWS * CSTR)     // [112][CSTR]
#define LDS_YW (LDS_Y + YROWS * CSTR)     // [9][CSTR]
#define LDS_TOTAL (LDS_YW + 9 * CSTR)     // 47256 floats = 189024 B (< 320KB WGP LDS)

// ---- compile-time monomial tables -----------------------------------------
struct Tab3 { unsigned char p[NP3], q[NP3], r[NP3], mult[NP3]; };
static constexpr Tab3 mk3() {
  Tab3 T{};
  int i = 0;
  for (int a = 0; a < DIM; a++)
    for (int b = a; b < DIM; b++)
      for (int c = b; c < DIM; c++) {
        T.p[i] = (unsigned char)a; T.q[i] = (unsigned char)b; T.r[i] = (unsigned char)c;
        T.mult[i] = (unsigned char)((a == b && b == c) ? 1 : ((a == b || b == c) ? 3 : 6));
        i++;
      }
  return T;
}
__constant__ Tab3 G3 = mk3();

struct Tab2 { unsigned char p[NP2], q[NP2], mult[NP2]; };
static constexpr Tab2 mk2() {
  Tab2 T{};
  int i = 0;
  for (int a = 0; a < DIM; a++)
    for (int b = a; b < DIM; b++) {
      T.p[i] = (unsigned char)a; T.q[i] = (unsigned char)b;
      T.mult[i] = (unsigned char)((a == b) ? 1 : 2);
      i++;
    }
  return T;
}
__constant__ Tab2 G2 = mk2();

__device__ __forceinline__ void lm_decode(int lm, int& l, int& m) {
  if (lm == 0)      { l = 0; m = 0; }
  else if (lm < 4)  { l = 1; m = lm - 1; }
  else              { l = 2; m = lm - 4; }
}

// ---- prep: fold permutation multiplicities into padded U matrices ---------
__global__ void prep_u(const float* __restrict__ u1_0, const float* __restrict__ u2_0,
                       const float* __restrict__ u3_0,
                       const float* __restrict__ u1_1, const float* __restrict__ u2_1,
                       const float* __restrict__ u3_1,
                       const float* __restrict__ u1_2, const float* __restrict__ u2_2,
                       const float* __restrict__ u3_2,
                       float* __restrict__ U) {
  int idx = blockIdx.x * blockDim.x + threadIdx.x;
  if (idx >= U_TOTAL) return;
  float val = 0.f;
  if (idx < U2_OFF) {
    int row = idx / K3P, t = idx % K3P;
    if (row < 63 && t < NP3) {
      int lm = row / 7, k = row % 7, l, m;
      lm_decode(lm, l, m);
      const float* u3 = (l == 0) ? u3_0 : ((l == 1) ? u3_1 : u3_2);
      int p = G3.p[t], q = G3.q[t], r = G3.r[t];
      val = (float)G3.mult[t] * u3[(((m * DIM + p) * DIM + q) * DIM + r) * 7 + k];
    }
  } else if (idx < U1_OFF) {
    int li = idx - U2_OFF;
    int row = li / K2P, t = li % K2P;
    if (row < 27 && t < NP2) {
      int lm = row / 3, k = row % 3, l, m;
      lm_decode(lm, l, m);
      const float* u2 = (l == 0) ? u2_0 : ((l == 1) ? u2_1 : u2_2);
      int p = G2.p[t], q = G2.q[t];
      val = (float)G2.mult[t] * u2[((m * DIM + p) * DIM + q) * 3 + k];
    }
  } else {
    int li = idx - U1_OFF;
    int row = li / K1P, t = li % K1P;
    if (row < 9 && t < DIM) {
      int l, m;
      lm_decode(row, l, m);
      const float* u1 = (l == 0) ? u1_0 : ((l == 1) ? u1_1 : u1_2);
      val = u1[m * DIM + t];   // K1 == 1
    }
  }
  U[idx] = val;
}

// ---- main fused per-node kernel -------------------------------------------
__global__ __launch_bounds__(256) void mace_node_kernel(
    const float* __restrict__ feats, const int* __restrict__ specie,
    const float* __restrict__ w1_0, const float* __restrict__ w2_0, const float* __restrict__ w3_0,
    const float* __restrict__ w1_1, const float* __restrict__ w2_1, const float* __restrict__ w3_1,
    const float* __restrict__ w1_2, const float* __restrict__ w2_2, const float* __restrict__ w3_2,
    const float* __restrict__ wl_0, const float* __restrict__ wl_1, const float* __restrict__ wl_2,
    const float* __restrict__ U, float* __restrict__ out) {
  extern __shared__ float smem[];
  const int b = blockIdx.x;
  const int tid = threadIdx.x;
  const int lane = tid & 31;
  // force wave index into an SGPR: tile params / loop bounds become scalar
  const int wv = __builtin_amdgcn_readfirstlane(tid >> 5);
  const int half = lane >> 4;   // WMMA half-wave select
  const int l16 = lane & 15;

  // x transposed into LDS: xt[p][c]   (feats is [b][c][p])
  const float* xb = feats + (size_t)b * (C * DIM);
  for (int i = tid; i < C * DIM; i += 256) {
    int c = i / DIM, p = i % DIM;
    smem[LDS_XT + p * CSTR + c] = xb[i];
  }
  __syncthreads();

  // moment matrix M[row][c]  (rows: 0..164 triples, 168..212 pairs, 216..224 singles, rest 0)
  for (int i = tid; i < MROWS * C; i += 256) {
    int row = i / C, c = i % C;
    float v = 0.f;
    if (row < NP3) {
      v = smem[LDS_XT + G3.p[row] * CSTR + c] *
          smem[LDS_XT + G3.q[row] * CSTR + c] *
          smem[LDS_XT + G3.r[row] * CSTR + c];
    } else if (row >= K3P && row < K3P + NP2) {
      int t = row - K3P;
      v = smem[LDS_XT + G2.p[t] * CSTR + c] * smem[LDS_XT + G2.q[t] * CSTR + c];
    } else if (row >= K3P + K2P && row < K3P + K2P + DIM) {
      v = smem[LDS_XT + (row - (K3P + K2P)) * CSTR + c];
    }
    smem[LDS_M + row * CSTR + c] = v;
  }
  __syncthreads();

  // GEMM  Y = U * M  via V_WMMA_F32_16X16X4_F32
  // 7 row tiles (4 nu3, 2 nu2, 1 nu1) x 8 col tiles, distributed over 8 waves.
  // Inner k-loop is software-pipelined: next fragments load during current WMMA.
  for (int tile = wv; tile < 56; tile += 8) {
    const int rt = tile >> 3, ct = tile & 7;
    const float* A; int Kp, ksteps, brow0, yrow0;
    if (rt < 4)      { A = U + rt * 16 * K3P;                 Kp = K3P; ksteps = K3P / 4; brow0 = 0;         yrow0 = rt * 16; }
    else if (rt < 6) { A = U + U2_OFF + (rt - 4) * 16 * K2P;  Kp = K2P; ksteps = K2P / 4; brow0 = K3P;       yrow0 = 64 + (rt - 4) * 16; }
    else             { A = U + U1_OFF;                        Kp = K1P; ksteps = K1P / 4; brow0 = K3P + K2P; yrow0 = 96; }

    v8f acc = {0.f, 0.f, 0.f, 0.f, 0.f, 0.f, 0.f, 0.f};
    // A frag: lanes 0-15 -> rows 0-15 K={k0,k0+1}; lanes 16-31 -> K={k0+2,k0+3}
    const float* Arow = A + l16 * Kp + half * 2;
    // B frag: lanes 0-15 -> B[k0..k0+1][col]; lanes 16-31 -> B[k0+2..k0+3][col]
    const float* Bb = smem + LDS_M + (brow0 + half * 2) * CSTR + ct * 16 + l16;

    v2f a, bv;
    a.x  = Arow[0];
    a.y  = Arow[1];
    bv.x = Bb[0];
    bv.y = Bb[CSTR];
    for (int kt = 0; kt < ksteps - 1; ++kt) {
      v2f an, bn;
      an.x = Arow[(kt + 1) * 4 + 0];
      an.y = Arow[(kt + 1) * 4 + 1];
      bn.x = Bb[(kt + 1) * 4 * CSTR];
      bn.y = Bb[(kt + 1) * 4 * CSTR + CSTR];
      acc = __builtin_amdgcn_wmma_f32_16x16x4_f32(false, a, false, bv, (short)0, acc,
                                                  false, false);
      a = an; bv = bn;
    }
    acc = __builtin_amdgcn_wmma_f32_16x16x4_f32(false, a, false, bv, (short)0, acc,
                                                false, false);

    // D: vgpr i -> row i (lanes<16) / row i+8 (lanes>=16), col = ct*16 + l16
    float* Yt = smem + LDS_Y + (yrow0 + half * 8) * CSTR + ct * 16 + l16;
#pragma unroll
    for (int i = 0; i < 8; ++i) Yt[i * CSTR] = acc[i];
  }
  __syncthreads();

  // combine over CG-path index k with species-gathered weights
  const int spec = specie[b];
  for (int i = tid; i < 9 * C; i += 256) {
    int lm = i / C, c = i % C, l, m;
    lm_decode(lm, l, m);
    const float* w3 = ((l == 0) ? w3_0 : (l == 1) ? w3_1 : w3_2) + spec * 7 * C + c;
    const float* w2 = ((l == 0) ? w2_0 : (l == 1) ? w2_1 : w2_2) + spec * 3 * C + c;
    const float* w1 = ((l == 0) ? w1_0 : (l == 1) ? w1_1 : w1_2) + spec * 1 * C + c;
    float y = 0.f;
#pragma unroll
    for (int k = 0; k < 7; ++k) y += w3[k * C] * smem[LDS_Y + (lm * 7 + k) * CSTR + c];
#pragma unroll
    for (int k = 0; k < 3; ++k) y += w2[k * C] * smem[LDS_Y + (64 + lm * 3 + k) * CSTR + c];
    y += w1[0] * smem[LDS_Y + (96 + lm) * CSTR + c];
    smem[LDS_YW + lm * CSTR + c] = y;
  }
  __syncthreads();

  // equivariant linear per irrep via WMMA:
  // out[b, off_l + d*m_l + m] = (1/sqrt(C)) sum_c yw[lmb+m][c] * wl_l[c][d]
  // A = yw rows (m_l real rows, zero-padded to 16), K = 128 channels, B = wl_l.
  // 3 irreps x 8 col tiles = 24 tile jobs over 8 waves.
  const float scale = 0.08838834764831845f;  // 1/sqrt(128)
  float* ob = out + (size_t)b * 1152;
  for (int job = wv; job < 24; job += 8) {
    const int l = job >> 3, ct = job & 7;
    const int mL   = (l == 0) ? 1 : ((l == 1) ? 3 : 5);
    const int lmb  = (l == 0) ? 0 : ((l == 1) ? 1 : 4);
    const int offL = (l == 0) ? 0 : ((l == 1) ? 128 : 512);
    const float* wl = (l == 0) ? wl_0 : ((l == 1) ? wl_1 : wl_2);

    const bool rowok = (l16 < mL);
    const float* Aw = smem + LDS_YW + (lmb + (rowok ? l16 : 0)) * CSTR + half * 2;
    const float* Bw = wl + (half * 2) * C + ct * 16 + l16;

    v8f acc = {0.f, 0.f, 0.f, 0.f, 0.f, 0.f, 0.f, 0.f};
    v2f a, bv;
    a.x  = rowok ? Aw[0] : 0.f;
    a.y  = rowok ? Aw[1] : 0.f;
    bv.x = Bw[0];
    bv.y = Bw[C];
    for (int kt = 0; kt < (C / 4) - 1; ++kt) {
      v2f an, bn;
      an.x = rowok ? Aw[(kt + 1) * 4 + 0] : 0.f;
      an.y = rowok ? Aw[(kt + 1) * 4 + 1] : 0.f;
      bn.x = Bw[(kt + 1) * 4 * C];
      bn.y = Bw[(kt + 1) * 4 * C + C];
      acc = __builtin_amdgcn_wmma_f32_16x16x4_f32(false, a, false, bv, (short)0, acc,
                                                  false, false);
      a = an; bv = bn;
    }
    acc = __builtin_amdgcn_wmma_f32_16x16x4_f32(false, a, false, bv, (short)0, acc,
                                                false, false);

    const int d = ct * 16 + l16;
#pragma unroll
    for (int i = 0; i < 8; ++i) {
      int row = i + half * 8;   // = m index
      if (row < mL) ob[offL + d * mL + row] = acc[i] * scale;
    }
  }
}

extern "C" void kernel_launch(void* const* d_in, const int* in_sizes, int n_in,
                              void* d_out, int out_size, void* d_ws, size_t ws_size,
                              hipStream_t stream) {
  (void)in_sizes; (void)n_in; (void)out_size; (void)ws_size;
  const float* feats  = (const float*)d_in[0];
  const int*   specie = (const int*)d_in[1];
  const float* u1_0 = (const float*)d_in[2],  *u2_0 = (const float*)d_in[3],  *u3_0 = (const float*)d_in[4];
  const float* w1_0 = (const float*)d_in[5],  *w2_0 = (const float*)d_in[6],  *w3_0 = (const float*)d_in[7];
  const float* wl_0 = (const float*)d_in[8];
  const float* u1_1 = (const float*)d_in[9],  *u2_1 = (const float*)d_in[10], *u3_1 = (const float*)d_in[11];
  const float* w1_1 = (const float*)d_in[12], *w2_1 = (const float*)d_in[13], *w3_1 = (const float*)d_in[14];
  const float* wl_1 = (const float*)d_in[15];
  const float* u1_2 = (const float*)d_in[16], *u2_2 = (const float*)d_in[17], *u3_2 = (const float*)d_in[18];
  const float* w1_2 = (const float*)d_in[19], *w2_2 = (const float*)d_in[20], *w3_2 = (const float*)d_in[21];
  const float* wl_2 = (const float*)d_in[22];

  float* U = (float*)d_ws;

  prep_u<<<(U_TOTAL + 255) / 256, 256, 0, stream>>>(u1_0, u2_0, u3_0, u1_1, u2_1, u3_1,
                                                    u1_2, u2_2, u3_2, U);

  const size_t lds_bytes = (size_t)LDS_TOTAL * sizeof(float);
  hipFuncSetAttribute((const void*)mace_node_kernel,
                      hipFuncAttributeMaxDynamicSharedMemorySize, (int)lds_bytes);
  mace_node_kernel<<<NNODES, 256, lds_bytes, stream>>>(
      feats, specie, w1_0, w2_0, w3_0, w1_1, w2_1, w3_1, w1_2, w2_2, w3_2,
      wl_0, wl_1, wl_2, U, (float*)d_out);
}